// SGATLayer_35811437314195
// MI455X (gfx1250) — compile-verified
//
#include <hip/hip_runtime.h>
#include <cstdint>

#define N_NODES 50000
#define N_EDGES 800000
#define IN_CH 64
#define OUT_CH 32
#define HEADS 4
#define HC 128            // HEADS*OUT_CH
#define NEG_SLOPE 0.2f
#define EPSV 1e-16f

typedef __attribute__((ext_vector_type(2))) float v2f;
typedef __attribute__((ext_vector_type(4))) float v4f;
typedef __attribute__((ext_vector_type(8))) float v8f;

// ---------------------------------------------------------------------------
// Kernel A: xh = x @ W.T  ([50000,64] x [64,128]) via V_WMMA_F32_16X16X4_F32.
// grid.x = 3125 row tiles of 16 nodes; block = 128 threads = 4 waves.
// Wave w owns output column tiles n0 = 32w and 32w+16 (two v8f accumulators).
// fp32 A 16x4 layout: lanes 0-15 hold K={k0,k0+1}, lanes 16-31 K={k0+2,k0+3};
// B 4x16 mirrored per column; C VGPR v -> rows v (lanes 0-15) / v+8 (16-31).
// ---------------------------------------------------------------------------
__global__ void __launch_bounds__(128) proj_nodes_wmma(
    const float* __restrict__ x, const float* __restrict__ W,
    float* __restrict__ xh)
{
    const int lane = threadIdx.x & 31;
    const int wv   = threadIdx.x >> 5;        // wave 0..3
    const int half = lane >> 4;               // 0 | 1
    const int l15  = lane & 15;
    const int m0   = blockIdx.x * 16;

    const int rowA  = m0 + l15;
    const int colB0 = wv * 32 + l15;
    const int colB1 = colB0 + 16;

    v8f acc0 = {};
    v8f acc1 = {};
#pragma unroll
    for (int k0 = 0; k0 < IN_CH; k0 += 4) {
        const int kk = k0 + half * 2;
        v2f a  = *(const v2f*)(x + (size_t)rowA  * IN_CH + kk);
        v2f b0 = *(const v2f*)(W + (size_t)colB0 * IN_CH + kk);
        v2f b1 = *(const v2f*)(W + (size_t)colB1 * IN_CH + kk);
        acc0 = __builtin_amdgcn_wmma_f32_16x16x4_f32(
                   false, a, false, b0, (short)0, acc0, false, false);
        acc1 = __builtin_amdgcn_wmma_f32_16x16x4_f32(
                   false, a, false, b1, (short)0, acc1, false, false);
    }
#pragma unroll
    for (int v = 0; v < 8; ++v) {
        const int m = m0 + v + half * 8;
        xh[(size_t)m * HC + colB0] = acc0[v];
        xh[(size_t)m * HC + colB1] = acc1[v];
    }
}

// ---------------------------------------------------------------------------
// w_eff[h,k] = sum_c att_edge[h,c] * W_edge[(h*32+c), k]   (4x64, 1 block)
// ---------------------------------------------------------------------------
__global__ void build_weff(const float* __restrict__ W_edge,
                           const float* __restrict__ att_edge,
                           float* __restrict__ weff)
{
    const int t = threadIdx.x;                 // 256 = HEADS*IN_CH
    const int h = t >> 6, k = t & 63;
    float s = 0.f;
#pragma unroll
    for (int c = 0; c < OUT_CH; ++c)
        s += att_edge[h * OUT_CH + c] * W_edge[(size_t)(h * OUT_CH + c) * IN_CH + k];
    weff[t] = s;
}

// ---------------------------------------------------------------------------
// Per-node attention scalars: s_node[n,h]=xh[n,h,:]·att_src[h], d_node with att_dst
// ---------------------------------------------------------------------------
__global__ void node_attn_scalars(const float* __restrict__ xh,
                                  const float* __restrict__ att_src,
                                  const float* __restrict__ att_dst,
                                  float* __restrict__ s_node,
                                  float* __restrict__ d_node)
{
    const int t = blockIdx.x * blockDim.x + threadIdx.x;
    if (t >= N_NODES * HEADS) return;
    const int n = t >> 2, h = t & 3;
    const float* xp = xh + (size_t)n * HC + h * OUT_CH;
    float ss = 0.f, dd = 0.f;
#pragma unroll
    for (int c = 0; c < OUT_CH; ++c) {
        const float v = xp[c];
        ss += v * att_src[h * OUT_CH + c];
        dd += v * att_dst[h * OUT_CH + c];
    }
    s_node[t] = ss;
    d_node[t] = dd;
}

// ---------------------------------------------------------------------------
// Zero/neg-inf init of accumulation buffers (d_ws is poisoned by harness).
// ---------------------------------------------------------------------------
__global__ void init_buffers(float* __restrict__ seg_max,
                             float* __restrict__ seg_sum,
                             float* __restrict__ acc)
{
    const int t = blockIdx.x * blockDim.x + threadIdx.x;
    if (t < N_NODES * HEADS) {
        seg_max[t] = -__builtin_huge_valf();
        seg_sum[t] = 0.f;
    }
    if (t < N_NODES * OUT_CH) acc[t] = 0.f;
}

// ---------------------------------------------------------------------------
// Edge pass 1: raw logits + leaky_relu + atomic segment max over dst.
// One thread per edge. edge_attr is 205MB read exactly once: load it with
// the non-temporal hint so it does not evict the reused working set
// (xh / acc / seg_* / alpha, all < 50MB total) from the 192MB L2.
// ---------------------------------------------------------------------------
__global__ void __launch_bounds__(256) edge_logits(
    const float* __restrict__ edge_attr, const int* __restrict__ edge_index,
    const float* __restrict__ s_node, const float* __restrict__ d_node,
    const float* __restrict__ weff,
    float* __restrict__ alpha, float* __restrict__ seg_max)
{
    __shared__ float sw[HEADS * IN_CH];
    if (threadIdx.x < HEADS * IN_CH) sw[threadIdx.x] = weff[threadIdx.x];
    __syncthreads();

    const int e = blockIdx.x * blockDim.x + threadIdx.x;
    if (e >= N_EDGES) return;
    const int src = edge_index[e];
    const int dst = edge_index[N_EDGES + e];

    float acch[HEADS] = {0.f, 0.f, 0.f, 0.f};
    const v4f* ea = (const v4f*)(edge_attr + (size_t)e * IN_CH);
#pragma unroll
    for (int k4 = 0; k4 < IN_CH / 4; ++k4) {
        const v4f v = __builtin_nontemporal_load(ea + k4);   // TH=NT stream
#pragma unroll
        for (int h = 0; h < HEADS; ++h) {
            const float* wh = sw + h * IN_CH + k4 * 4;
            acch[h] += v.x * wh[0] + v.y * wh[1] + v.z * wh[2] + v.w * wh[3];
        }
    }
#pragma unroll
    for (int h = 0; h < HEADS; ++h) {
        float a = acch[h] + s_node[dst * 4 + h] + d_node[src * 4 + h];
        a = (a > 0.f) ? a : NEG_SLOPE * a;
        alpha[(size_t)e * 4 + h] = a;
        float* addr = seg_max + dst * 4 + h;   // IEEE monotone int trick
        if (a >= 0.f) atomicMax((int*)addr, __float_as_int(a));
        else          atomicMin((unsigned int*)addr, __float_as_uint(a));
    }
}

// ---------------------------------------------------------------------------
// Edge pass 2: alpha <- exp(alpha - segmax[dst]); seg_sum[dst] += alpha.
// ---------------------------------------------------------------------------
__global__ void edge_softmax(const int* __restrict__ edge_index,
                             const float* __restrict__ seg_max,
                             float* __restrict__ alpha,
                             float* __restrict__ seg_sum)
{
    const int t = blockIdx.x * blockDim.x + threadIdx.x;
    if (t >= N_EDGES * HEADS) return;
    const int e = t >> 2, h = t & 3;
    const int dst = edge_index[N_EDGES + e];
    const float ex = __expf(alpha[t] - seg_max[dst * 4 + h]);
    alpha[t] = ex;
    unsafeAtomicAdd(seg_sum + dst * 4 + h, ex);   // global_atomic_add_f32 @ L2
}

// ---------------------------------------------------------------------------
// Edge pass 3: acc[dst,c] += sum_h coef[e,h] * xh[src,h*32+c].
// One wave (32 lanes = 32 channels) per edge; xh + acc are L2-resident.
// ---------------------------------------------------------------------------
__global__ void __launch_bounds__(256) edge_scatter(
    const int* __restrict__ edge_index, const float* __restrict__ xh,
    const float* __restrict__ alpha, const float* __restrict__ seg_sum,
    float* __restrict__ acc)
{
    const int lane = threadIdx.x & 31;
    const int e = blockIdx.x * 8 + (threadIdx.x >> 5);
    if (e >= N_EDGES) return;
    const int src = edge_index[e];
    const int dst = edge_index[N_EDGES + e];

    const float c0 = alpha[(size_t)e * 4 + 0] / (seg_sum[dst * 4 + 0] + EPSV);
    const float c1 = alpha[(size_t)e * 4 + 1] / (seg_sum[dst * 4 + 1] + EPSV);
    const float c2 = alpha[(size_t)e * 4 + 2] / (seg_sum[dst * 4 + 2] + EPSV);
    const float c3 = alpha[(size_t)e * 4 + 3] / (seg_sum[dst * 4 + 3] + EPSV);

    const float* xp = xh + (size_t)src * HC;
    const float v = c0 * xp[lane] + c1 * xp[32 + lane]
                  + c2 * xp[64 + lane] + c3 * xp[96 + lane];
    unsafeAtomicAdd(acc + (size_t)dst * OUT_CH + lane, v);
}

// ---------------------------------------------------------------------------
// Finalize: out = relu(acc/HEADS + bias_conv + bias_layer)
// ---------------------------------------------------------------------------
__global__ void finalize(const float* __restrict__ acc,
                         const float* __restrict__ bias_conv,
                         const float* __restrict__ bias_layer,
                         float* __restrict__ out)
{
    const int t = blockIdx.x * blockDim.x + threadIdx.x;
    if (t >= N_NODES * OUT_CH) return;
    const int c = t & 31;
    const float v = acc[t] * 0.25f + bias_conv[c] + bias_layer[c];
    out[t] = v > 0.f ? v : 0.f;
}

// ---------------------------------------------------------------------------
extern "C" void kernel_launch(void* const* d_in, const int* in_sizes, int n_in,
                              void* d_out, int out_size, void* d_ws, size_t ws_size,
                              hipStream_t stream) {
    const float* x          = (const float*)d_in[0];
    const float* edge_attr  = (const float*)d_in[1];
    const float* W          = (const float*)d_in[2];
    const float* W_edge     = (const float*)d_in[3];
    const float* att_src    = (const float*)d_in[4];
    const float* att_dst    = (const float*)d_in[5];
    const float* att_edge   = (const float*)d_in[6];
    const float* bias_conv  = (const float*)d_in[7];
    const float* bias_layer = (const float*)d_in[8];
    const int*   edge_index = (const int*)d_in[9];
    float* out = (float*)d_out;

    // Workspace layout (floats), total ~48.0 MB
    float* ws      = (float*)d_ws;
    float* xh      = ws;                         // 50000*128 = 6,400,000
    float* s_node  = xh      + (size_t)N_NODES * HC;      // 200,000
    float* d_node  = s_node  + (size_t)N_NODES * HEADS;   // 200,000
    float* weff    = d_node  + (size_t)N_NODES * HEADS;   // 256
    float* alpha   = weff    + (size_t)HEADS * IN_CH;     // 3,200,000
    float* seg_max = alpha   + (size_t)N_EDGES * HEADS;   // 200,000
    float* seg_sum = seg_max + (size_t)N_NODES * HEADS;   // 200,000
    float* accb    = seg_sum + (size_t)N_NODES * HEADS;   // 1,600,000

    proj_nodes_wmma<<<N_NODES / 16, 128, 0, stream>>>(x, W, xh);
    build_weff<<<1, 256, 0, stream>>>(W_edge, att_edge, weff);
    node_attn_scalars<<<(N_NODES * HEADS + 255) / 256, 256, 0, stream>>>(
        xh, att_src, att_dst, s_node, d_node);
    init_buffers<<<(N_NODES * OUT_CH + 255) / 256, 256, 0, stream>>>(
        seg_max, seg_sum, accb);
    edge_logits<<<(N_EDGES + 255) / 256, 256, 0, stream>>>(
        edge_attr, edge_index, s_node, d_node, weff, alpha, seg_max);
    edge_softmax<<<(N_EDGES * HEADS + 255) / 256, 256, 0, stream>>>(
        edge_index, seg_max, alpha, seg_sum);
    edge_scatter<<<(N_EDGES + 7) / 8, 256, 0, stream>>>(
        edge_index, xh, alpha, seg_sum, accb);
    finalize<<<(N_NODES * OUT_CH + 255) / 256, 256, 0, stream>>>(
        accb, bias_conv, bias_layer, out);
}